// DynamicGraphConvolutionKerasLayer_34222299414788
// MI455X (gfx1250) — compile-verified
//
#include <hip/hip_runtime.h>

typedef float v2f __attribute__((ext_vector_type(2)));
typedef float v8f __attribute__((ext_vector_type(8)));

#define VV    50000
#define CC    128
#define CO    256
#define DEG   16
#define K2DIM 256          // 2*CC
#define STAGES 16          // K stages of 16 rows each
#define STK    16          // K rows per stage
#define KPST   8           // K-pairs per stage
#define PITCHW 544         // dwords per K-pair row in LDS (512 + 32 pad -> bank-split halves)

// ---------------------------------------------------------------------------
// Kernel 1: fold the concat-linear into one 256x256 matrix, stored K-pair
// interleaved:  k2g[p*512 + col*2 + r] = Kfold[2p + r][col]
//   Kfold[0:128]   = K1 - K2
//   Kfold[128:256] = K2
// This makes the WMMA B fragment (rows k,k+1 at one column) a contiguous 8B.
// ---------------------------------------------------------------------------
__global__ void prep_kernel(const float* __restrict__ kin, float* __restrict__ k2g) {
    int n = threadIdx.x;          // column 0..255
    int k = blockIdx.x;           // row    0..255
    float v = kin[k * CO + n];
    if (k < CC) v -= kin[(k + CC) * CO + n];
    k2g[(size_t)(k >> 1) * (2 * CO) + n * 2 + (k & 1)] = v;
}

// ---------------------------------------------------------------------------
// Kernel 2: per-vertex weighted neighbor aggregation (segments contiguous:
// edge_src == repeat(arange(V),16)).  One wave per vertex, lane owns 4 chans.
// feat2[i] = [s_i * x_i , ybar_i],  svec[i] = s_i.
// ---------------------------------------------------------------------------
__global__ __launch_bounds__(256) void agg_kernel(
    const float* __restrict__ data, const int* __restrict__ edge_dst,
    const float* __restrict__ edge_w, float* __restrict__ feat2,
    float* __restrict__ svec) {
    int wave = threadIdx.x >> 5;
    int lane = threadIdx.x & 31;
    int i = blockIdx.x * 8 + wave;
    if (i >= VV) return;

    const float4* d4 = (const float4*)data;          // [VV][CC/4]
    float4 acc = make_float4(0.f, 0.f, 0.f, 0.f);
    float s = 0.f;
    int ebase = i * DEG;
    #pragma unroll 4
    for (int e = 0; e < DEG; ++e) {
        int   d = edge_dst[ebase + e];               // wave-uniform -> scalar
        float w = edge_w[ebase + e];
        float4 y = d4[d * (CC / 4) + lane];          // L2-resident gather
        acc.x += w * y.x; acc.y += w * y.y;
        acc.z += w * y.z; acc.w += w * y.w;
        s += w;
    }
    float4 x = d4[i * (CC / 4) + lane];
    float4* f4 = (float4*)feat2;                     // [VV][2*CC/4]
    f4[i * (2 * CC / 4) + lane] = make_float4(s * x.x, s * x.y, s * x.z, s * x.w);
    f4[i * (2 * CC / 4) + (CC / 4) + lane] = acc;
    if (lane == 0) svec[i] = s;
}

// ---------------------------------------------------------------------------
// Kernel 3: out = feat2 @ Kfold + s*bias via V_WMMA_F32_16X16X4_F32.
// 8 waves/block, each wave owns a 16x256 strip (16 v8f accumulators).
// B double-buffered in LDS, filled by async global->LDS DMA (ASYNCcnt),
// A fragments software-pipelined in registers.
// ---------------------------------------------------------------------------
__global__ __launch_bounds__(256) void gemm_kernel(
    const float* __restrict__ feat2, const float* __restrict__ k2g,
    const float* __restrict__ svec, const float* __restrict__ bias,
    float* __restrict__ out) {
    __shared__ __align__(16) float bsh[2][KPST * PITCHW];   // 2 x 17408 B

    const int tid  = threadIdx.x;
    const int wave = tid >> 5;
    const int lane = tid & 31;
    const int mt   = blockIdx.x * 8 + wave;
    const int m0   = mt * 16;
    const bool valid = (mt < VV / 16);               // wave-uniform

    const int halfsel = lane >> 4;                   // 0: lanes 0-15, 1: 16-31
    const int koff    = halfsel * 2;                 // A/B fragment K offset
    const int l15     = lane & 15;
    const int arow    = (valid ? m0 : 0) + l15;
    const float* arowp = feat2 + (size_t)arow * K2DIM + koff;

    // Async B copy: this wave DMAs K-pair row `wave`, 4 quarters of 512B.
    const uint32_t lds0 = (uint32_t)(uintptr_t)(&bsh[0][0]);
    const uint32_t lds1 = (uint32_t)(uintptr_t)(&bsh[1][0]);
    const uint32_t lds_lane = (uint32_t)(wave * (PITCHW * 4) + lane * 16);
    const char* gsrc_lane = (const char*)k2g + (size_t)wave * 2048 + (size_t)lane * 16;

    auto issue_stage = [&](int st, int buf) {
        uint32_t ldsb = (buf ? lds1 : lds0) + lds_lane;
        const char* g = gsrc_lane + (size_t)st * (KPST * 2048);
        #pragma unroll
        for (int q = 0; q < 4; ++q) {
            uint32_t    d = ldsb + q * 512;
            const char* s = g + q * 512;
            asm volatile("global_load_async_to_lds_b128 %0, %1, off"
                         :: "v"(d), "v"(s) : "memory");
        }
    };

    v8f acc[16] = {};
    v2f acur[4] = {}, anxt[4] = {};

    issue_stage(0, 0);
    if (valid) {
        #pragma unroll
        for (int i = 0; i < 4; ++i) acur[i] = *(const v2f*)(arowp + i * 4);
    }
    asm volatile("s_wait_asynccnt 0x0" ::: "memory");
    __syncthreads();

    for (int st = 0; st < STAGES; ++st) {
        if (st + 1 < STAGES) {
            issue_stage(st + 1, (st + 1) & 1);        // overlap DMA with compute
            if (valid) {
                #pragma unroll
                for (int i = 0; i < 4; ++i)
                    anxt[i] = *(const v2f*)(arowp + (st + 1) * STK + i * 4);
            }
        }
        if (valid) {
            const float* bb = &bsh[st & 1][0];
            #pragma unroll
            for (int k4 = 0; k4 < 4; ++k4) {
                const float* brow = bb + (size_t)(k4 * 2 + halfsel) * PITCHW + l15 * 2;
                v2f a = acur[k4];
                #pragma unroll
                for (int nt = 0; nt < 16; ++nt) {
                    v2f b = *(const v2f*)(brow + nt * 32);   // one ds_load_b64
                    acc[nt] = __builtin_amdgcn_wmma_f32_16x16x4_f32(
                        false, a, false, b, (short)0, acc[nt], false, false);
                }
            }
        }
        if (st + 1 < STAGES) {
            asm volatile("s_wait_asynccnt 0x0" ::: "memory");
            __syncthreads();
            #pragma unroll
            for (int i = 0; i < 4; ++i) acur[i] = anxt[i];
        }
    }

    if (valid) {
        #pragma unroll
        for (int nt = 0; nt < 16; ++nt) {
            int col = nt * 16 + l15;
            float bcol = bias[col];
            #pragma unroll
            for (int r = 0; r < 8; ++r) {
                int row = m0 + r + halfsel * 8;       // C/D layout: VGPR r -> M
                out[(size_t)row * CO + col] = acc[nt][r] + svec[row] * bcol;
            }
        }
    }
}

// ---------------------------------------------------------------------------
extern "C" void kernel_launch(void* const* d_in, const int* in_sizes, int n_in,
                              void* d_out, int out_size, void* d_ws, size_t ws_size,
                              hipStream_t stream) {
    const float* data     = (const float*)d_in[0];
    // d_in[1] = edge_src: unused (segments are contiguous by construction)
    const int*   edge_dst = (const int*)d_in[2];
    const float* edge_w   = (const float*)d_in[3];
    const float* kin      = (const float*)d_in[4];
    const float* bias     = (const float*)d_in[5];
    float* out = (float*)d_out;

    float* ws    = (float*)d_ws;
    float* feat2 = ws;                                   // VV * 256 floats
    float* k2g   = feat2 + (size_t)VV * K2DIM;           // 256 * 256 floats (interleaved)
    float* svec  = k2g + (size_t)K2DIM * CO;             // VV floats

    prep_kernel<<<CO, CO, 0, stream>>>(kin, k2g);
    agg_kernel<<<(VV + 7) / 8, 256, 0, stream>>>(data, edge_dst, edge_w, feat2, svec);
    gemm_kernel<<<((VV / 16) + 7) / 8, 256, 0, stream>>>(feat2, k2g, svec, bias, out);
}